// patch_PSUB_47545287966795
// MI455X (gfx1250) — compile-verified
//
#include <hip/hip_runtime.h>

typedef __attribute__((ext_vector_type(2))) float v2f;
typedef __attribute__((ext_vector_type(8))) float v8f;

// ---------------------------------------------------------------------------
// Y[b,m,:64] = s * (L @ X_b)[m,:] - (use_z ? Z[b,m,:] : 0)
// L: [V,V] row-major, X/Z/Y: [B,V,64]. Grid: (V/64, B), block 128 (4 waves).
// Each wave computes a 16x64 strip with v_wmma_f32_16x16x4_f32.
// LDS layouts give each WMMA fragment half as one aligned ds_load_b64.
// ---------------------------------------------------------------------------
__global__ __launch_bounds__(128)
void cheb_lap_gemm_kernel(const float* __restrict__ L,
                          const float* __restrict__ X,
                          const float* __restrict__ Z,
                          float* __restrict__ Y,
                          int V, float s, int use_z)
{
    // A tile: LtQ[row][kk][j] = L[m0+row][k0+4*kk+j]; row stride 68 dwords
    // -> lane reads contiguous float2 at [row][kk][2*halfsel], banks 4*(row+kk)
    __shared__ float LtQ[64][17][4];
    // B tile: XtQ[kk][col][j] = X[k0+4*kk+j][col]; per-column 16B chunk
    // -> lanes l / l+16 read the two halves of one chunk, all 64 banks covered
    __shared__ float XtQ[16][64][4];

    const int tid  = threadIdx.x;
    const int wave = tid >> 5;
    const int lane = tid & 31;
    const int b    = blockIdx.y;
    const int m0   = blockIdx.x * 64;

    const float* __restrict__ Xb = X + (size_t)b * V * 64;

    v8f acc[4];
#pragma unroll
    for (int nt = 0; nt < 4; ++nt)
#pragma unroll
        for (int j = 0; j < 8; ++j) acc[nt][j] = 0.0f;

    const int halfsel = lane >> 4;   // 0: lanes 0-15 (K0,K1), 1: lanes 16-31 (K2,K3)
    const int l16     = lane & 15;
    const int h2      = halfsel * 2;
    const int arow    = wave * 16 + l16;

    for (int k0 = 0; k0 < V; k0 += 64) {
        __syncthreads();
        // stage L tile: rows m0..m0+63, cols k0..k0+63 (one b128 LDS store each)
#pragma unroll
        for (int p = 0; p < 8; ++p) {
            int r = p * 8 + (tid >> 4);
            int c = (tid & 15) << 2;
            float4 lv = *reinterpret_cast<const float4*>(&L[(size_t)(m0 + r) * V + (k0 + c)]);
            *reinterpret_cast<float4*>(&LtQ[r][c >> 2][0]) = lv;
        }
        // stage X tile: rows k0..k0+63, cols 0..63, K-grouped scatter
#pragma unroll
        for (int p = 0; p < 8; ++p) {
            int kr = p * 8 + (tid >> 4);
            int kc = (tid & 15) << 2;
            float4 xv = *reinterpret_cast<const float4*>(&Xb[(size_t)(k0 + kr) * 64 + kc]);
            XtQ[kr >> 2][kc + 0][kr & 3] = xv.x;
            XtQ[kr >> 2][kc + 1][kr & 3] = xv.y;
            XtQ[kr >> 2][kc + 2][kr & 3] = xv.z;
            XtQ[kr >> 2][kc + 3][kr & 3] = xv.w;
        }
        __syncthreads();

#pragma unroll
        for (int kk = 0; kk < 16; ++kk) {
            v2f a = *reinterpret_cast<const v2f*>(&LtQ[arow][kk][h2]);
#pragma unroll
            for (int nt = 0; nt < 4; ++nt) {
                v2f bf = *reinterpret_cast<const v2f*>(&XtQ[kk][nt * 16 + l16][h2]);
                acc[nt] = __builtin_amdgcn_wmma_f32_16x16x4_f32(
                    false, a, false, bf, (short)0, acc[nt], false, false);
            }
        }
    }

    float* __restrict__ Yb = Y + (size_t)b * V * 64;
    const float* __restrict__ Zb = Z + (size_t)b * V * 64;
#pragma unroll
    for (int nt = 0; nt < 4; ++nt) {
#pragma unroll
        for (int j = 0; j < 8; ++j) {
            int row = m0 + wave * 16 + j + halfsel * 8;  // C/D: VGPR j -> rows j / j+8
            int col = nt * 16 + l16;
            float v = s * acc[nt][j];
            if (use_z) v -= Zb[(size_t)row * 64 + col];
            Yb[(size_t)row * 64 + col] = v;
        }
    }
}

// ---------------------------------------------------------------------------
// H[b,m,n0+n] (+)= (T_b @ Wk)[m,n], K=64. init_from_bias: C starts from bias.
// T: [B,V,64], Wk: [64,Nout], H: [B,V,Nout]. Grid (V/64, Nout/64, B), block 128.
// ---------------------------------------------------------------------------
__global__ __launch_bounds__(128)
void cheb_accum_w_kernel(const float* __restrict__ T,
                         const float* __restrict__ Wk,
                         const float* __restrict__ bias,
                         float* __restrict__ H,
                         int V, int Nout, int init_from_bias)
{
    __shared__ float TtQ[64][17][4];
    __shared__ float WtQ[16][64][4];

    const int tid  = threadIdx.x;
    const int wave = tid >> 5;
    const int lane = tid & 31;
    const int m0   = blockIdx.x * 64;
    const int n0   = blockIdx.y * 64;
    const int b    = blockIdx.z;

    const float* __restrict__ Tb = T + (size_t)b * V * 64;
    float* __restrict__ Hb = H + (size_t)b * V * Nout;

    const int halfsel = lane >> 4;
    const int l16     = lane & 15;
    const int h2      = halfsel * 2;
    const int arow    = wave * 16 + l16;

#pragma unroll
    for (int p = 0; p < 8; ++p) {
        int r = p * 8 + (tid >> 4);
        int c = (tid & 15) << 2;
        float4 tv = *reinterpret_cast<const float4*>(&Tb[(size_t)(m0 + r) * 64 + c]);
        *reinterpret_cast<float4*>(&TtQ[r][c >> 2][0]) = tv;
        float4 wv = *reinterpret_cast<const float4*>(&Wk[(size_t)r * Nout + (n0 + c)]);
        WtQ[r >> 2][c + 0][r & 3] = wv.x;
        WtQ[r >> 2][c + 1][r & 3] = wv.y;
        WtQ[r >> 2][c + 2][r & 3] = wv.z;
        WtQ[r >> 2][c + 3][r & 3] = wv.w;
    }

    v8f acc[4];
#pragma unroll
    for (int nt = 0; nt < 4; ++nt) {
#pragma unroll
        for (int j = 0; j < 8; ++j) {
            int row = m0 + wave * 16 + j + halfsel * 8;
            int col = n0 + nt * 16 + l16;
            acc[nt][j] = init_from_bias ? bias[col] : Hb[(size_t)row * Nout + col];
        }
    }
    __syncthreads();

#pragma unroll
    for (int kk = 0; kk < 16; ++kk) {
        v2f a = *reinterpret_cast<const v2f*>(&TtQ[arow][kk][h2]);
#pragma unroll
        for (int nt = 0; nt < 4; ++nt) {
            v2f bf = *reinterpret_cast<const v2f*>(&WtQ[kk][nt * 16 + l16][h2]);
            acc[nt] = __builtin_amdgcn_wmma_f32_16x16x4_f32(
                false, a, false, bf, (short)0, acc[nt], false, false);
        }
    }

#pragma unroll
    for (int nt = 0; nt < 4; ++nt) {
#pragma unroll
        for (int j = 0; j < 8; ++j) {
            int row = m0 + wave * 16 + j + halfsel * 8;
            int col = n0 + nt * 16 + l16;
            Hb[(size_t)row * Nout + col] = acc[nt][j];
        }
    }
}

// ---------------------------------------------------------------------------
// Hs[b, nb[j*4+q], o] = H[b, j, q*64+o]
// ---------------------------------------------------------------------------
__global__ void pixelshuffle_kernel(const float* __restrict__ H,
                                    const int* __restrict__ nb,
                                    float* __restrict__ Hs,
                                    int Vl, int B)
{
    int idx = blockIdx.x * blockDim.x + threadIdx.x;
    int total = B * Vl * 4 * 64;
    if (idx >= total) return;
    int o  = idx & 63;
    int t  = idx >> 6;        // b*Vl*4 + j*4 + q
    int q  = t & 3;
    int jj = t >> 2;
    int j  = jj % Vl;
    int b  = jj / Vl;
    int row = nb[j * 4 + q];
    Hs[((size_t)b * (4 * Vl) + row) * 64 + o] =
        H[((size_t)b * Vl + j) * 256 + (size_t)q * 64 + o];
}

extern "C" void kernel_launch(void* const* d_in, const int* in_sizes, int n_in,
                              void* d_out, int out_size, void* d_ws, size_t ws_size,
                              hipStream_t stream)
{
    (void)in_sizes; (void)n_in; (void)out_size; (void)ws_size;

    const float* x     = (const float*)d_in[0];   // [4,1024,64]
    const float* lap0  = (const float*)d_in[1];   // [1024,1024]
    const float* lap1  = (const float*)d_in[2];   // [4096,4096]
    const float* W_in  = (const float*)d_in[3];   // [5,64,256]
    const float* b_in  = (const float*)d_in[4];   // [256]
    const float* W_out = (const float*)d_in[5];   // [5,64,64]
    const float* b_out = (const float*)d_in[6];   // [64]
    const int*   nb    = (const int*)d_in[7];     // [1024,4]

    const int B = 4, Vl = 1024, nf = 64;
    const int Vh = 4 * Vl;

    float* H1 = (float*)d_ws;                       // [B,Vl,256]  = 4 MB
    float* Hs = H1 + (size_t)B * Vl * 4 * nf;       // [B,Vh,64]   = 4 MB
    float* Ta = Hs + (size_t)B * Vh * nf;           // [B,Vh,64]   = 4 MB
    float* Tb = Ta + (size_t)B * Vh * nf;           // [B,Vh,64]   = 4 MB
    float* out = (float*)d_out;                     // [B,Vh,64]

    dim3 blk(128);

    // ---------------- phase 1: Cheb(L0), accumulate H1 = sum_k T_k W_in[k] + b_in
    {
        dim3 gA(Vl / 64, 256 / 64, B);
        dim3 gG(Vl / 64, B);
        // k=0: T0 = x
        cheb_accum_w_kernel<<<gA, blk, 0, stream>>>(x, W_in + 0 * 64 * 256, b_in, H1, Vl, 256, 1);
        // T1 = L0 @ x
        cheb_lap_gemm_kernel<<<gG, blk, 0, stream>>>(lap0, x, Ta, Ta, Vl, 1.0f, 0);
        cheb_accum_w_kernel<<<gA, blk, 0, stream>>>(Ta, W_in + 1 * 64 * 256, b_in, H1, Vl, 256, 0);
        // T2 = 2 L0 T1 - T0
        cheb_lap_gemm_kernel<<<gG, blk, 0, stream>>>(lap0, Ta, x, Tb, Vl, 2.0f, 1);
        cheb_accum_w_kernel<<<gA, blk, 0, stream>>>(Tb, W_in + 2 * 64 * 256, b_in, H1, Vl, 256, 0);
        // T3 = 2 L0 T2 - T1   (in-place over T1 is safe: per-element read-then-write)
        cheb_lap_gemm_kernel<<<gG, blk, 0, stream>>>(lap0, Tb, Ta, Ta, Vl, 2.0f, 1);
        cheb_accum_w_kernel<<<gA, blk, 0, stream>>>(Ta, W_in + 3 * 64 * 256, b_in, H1, Vl, 256, 0);
        // T4 = 2 L0 T3 - T2
        cheb_lap_gemm_kernel<<<gG, blk, 0, stream>>>(lap0, Ta, Tb, Tb, Vl, 2.0f, 1);
        cheb_accum_w_kernel<<<gA, blk, 0, stream>>>(Tb, W_in + 4 * 64 * 256, b_in, H1, Vl, 256, 0);
    }

    // ---------------- pixel shuffle scatter
    {
        int total = B * Vl * 4 * nf;
        pixelshuffle_kernel<<<(total + 255) / 256, 256, 0, stream>>>(H1, nb, Hs, Vl, B);
    }

    // ---------------- phase 2: Cheb(L1), accumulate out = sum_k T_k W_out[k] + b_out
    {
        dim3 gA(Vh / 64, nf / 64, B);
        dim3 gG(Vh / 64, B);
        cheb_accum_w_kernel<<<gA, blk, 0, stream>>>(Hs, W_out + 0 * 64 * 64, b_out, out, Vh, nf, 1);
        cheb_lap_gemm_kernel<<<gG, blk, 0, stream>>>(lap1, Hs, Ta, Ta, Vh, 1.0f, 0);
        cheb_accum_w_kernel<<<gA, blk, 0, stream>>>(Ta, W_out + 1 * 64 * 64, b_out, out, Vh, nf, 0);
        cheb_lap_gemm_kernel<<<gG, blk, 0, stream>>>(lap1, Ta, Hs, Tb, Vh, 2.0f, 1);
        cheb_accum_w_kernel<<<gA, blk, 0, stream>>>(Tb, W_out + 2 * 64 * 64, b_out, out, Vh, nf, 0);
        cheb_lap_gemm_kernel<<<gG, blk, 0, stream>>>(lap1, Tb, Ta, Ta, Vh, 2.0f, 1);
        cheb_accum_w_kernel<<<gA, blk, 0, stream>>>(Ta, W_out + 3 * 64 * 64, b_out, out, Vh, nf, 0);
        cheb_lap_gemm_kernel<<<gG, blk, 0, stream>>>(lap1, Ta, Tb, Tb, Vh, 2.0f, 1);
        cheb_accum_w_kernel<<<gA, blk, 0, stream>>>(Tb, W_out + 4 * 64 * 64, b_out, out, Vh, nf, 0);
    }
}